// Mamba_20444044329721
// MI455X (gfx1250) — compile-verified
//
#include <hip/hip_runtime.h>
#include <math.h>

// MI455X / gfx1250: wave32, WMMA 16x16 shapes, 192MB L2 (whole problem L2-resident
// after first touch -> layout passes are cheap), HBM floor ~2.9us for 67MB.

typedef float v2f __attribute__((ext_vector_type(2)));
typedef float v8f __attribute__((ext_vector_type(8)));

#define BATCH 4
#define SEQL  4096
#define DCH   512
#define NST   16
#define CHUNK 16
#define NCHUNK (SEQL/CHUNK)   // 256 chunks per (b,d)
#define BT    (BATCH*SEQL)    // 16384

// ---------------------------------------------------------------------------
// Kernel A: per-channel tables, written directly in WMMA A-matrix lane layout.
//   M_d[t][s] = k_d[t-s] (t>=s), k_d[tau] = sum_n dB[n]*dA[n]^tau
//   P_d[t][n] = dA[n]^(t+1)
//   Q_d[n][s] = dA[n]^(15-s)*dB[n],  r_d[n] = dA[n]^16
// A-matrix (16x4 f32, ISA 7.12.2): lane<16 row=lane, VGPR v -> K = 2*(lane>>4)+v
// Stored as  T[((d*4+k)*32+lane)*2+v]  so each lane does one b64 load per K-slice.
__global__ void precompute_kernel(const float* __restrict__ A,
                                  const float* __restrict__ Bm,
                                  const float* __restrict__ delta,
                                  float* __restrict__ MA, float* __restrict__ PA,
                                  float* __restrict__ Q,  float* __restrict__ r)
{
    int d = blockIdx.x * blockDim.x + threadIdx.x;
    if (d >= DCH) return;
    float dt = delta[d];
    float Ad[NST], dBv[NST], dAv[NST];
#pragma unroll
    for (int n = 0; n < NST; n++) {
        Ad[n]  = A[d * NST + n];
        dBv[n] = dt * Bm[d * NST + n];
        dAv[n] = expf(dt * Ad[n]);
    }
    float kv[NST];
    {
        float p[NST];
#pragma unroll
        for (int n = 0; n < NST; n++) p[n] = dBv[n];
        for (int tau = 0; tau < NST; tau++) {
            float s = 0.f;
#pragma unroll
            for (int n = 0; n < NST; n++) s += p[n];
            kv[tau] = s;
#pragma unroll
            for (int n = 0; n < NST; n++) p[n] *= dAv[n];
        }
    }
    for (int k = 0; k < 4; k++)
        for (int lane = 0; lane < 32; lane++) {
            int row = lane & 15, hi = lane >> 4;
            for (int v = 0; v < 2; v++) {
                int kk = 4 * k + 2 * hi + v;
                MA[((d * 4 + k) * 32 + lane) * 2 + v] = (row >= kk) ? kv[row - kk] : 0.f;
                PA[((d * 4 + k) * 32 + lane) * 2 + v] = expf(dt * Ad[kk] * (float)(row + 1));
            }
        }
#pragma unroll
    for (int n = 0; n < NST; n++) {
        r[d * NST + n] = expf(16.f * dt * Ad[n]);
        for (int s = 0; s < CHUNK; s++)
            Q[d * 256 + n * 16 + s] = expf(dt * Ad[n] * (float)(15 - s)) * dBv[n];
    }
}

// ---------------------------------------------------------------------------
// LDS-tiled transpose:  in[R][C] -> out[C][R].   grid = (C/32, R/32), block (32,8)
__global__ void transpose_kernel(const float* __restrict__ in,
                                 float* __restrict__ out, int R, int C)
{
    __shared__ float tile[32][33];
    int c0 = blockIdx.x * 32, r0 = blockIdx.y * 32;
    int tx = threadIdx.x, ty = threadIdx.y;
#pragma unroll
    for (int i = 0; i < 32; i += 8)
        tile[ty + i][tx] = in[(size_t)(r0 + ty + i) * C + (c0 + tx)];
    __syncthreads();
#pragma unroll
    for (int i = 0; i < 32; i += 8)
        out[(size_t)(c0 + ty + i) * R + (r0 + tx)] = tile[tx][ty + i];
}

// ---------------------------------------------------------------------------
// Kernel C: per-(b,d,n) chunk-boundary states (the only sequential loop: 256 iters).
//   g_{c+1}[n] = r[n]*g_c[n] + sum_s Q[n][s] * x[chunk c][s];  H stores carry-IN g_c.
// Layout: XT/H at (d*4+b)*4096 + c*16 + {s|n}  (contiguous 64B per chunk).
__global__ void chunk_states_kernel(const float* __restrict__ XT,
                                    const float* __restrict__ Q,
                                    const float* __restrict__ r,
                                    float* __restrict__ H)
{
    int tid = blockIdx.x * blockDim.x + threadIdx.x;   // 32768 = B*D*N
    int n = tid & 15;
    int d = (tid >> 4) & (DCH - 1);
    int b = tid >> 13;
    const float* xb = XT + (size_t)(d * BATCH + b) * SEQL;
    float*       Hp = H  + (size_t)(d * BATCH + b) * SEQL;
    float q[16];
#pragma unroll
    for (int s = 0; s < 16; s++) q[s] = Q[d * 256 + n * 16 + s];
    float rn = r[d * NST + n];
    float h = 0.f;
    for (int c = 0; c < NCHUNK; c++) {
        Hp[c * 16 + n] = h;                       // carry-in for chunk c
        __builtin_prefetch(xb + (c + 4) * 16, 0, 3);
        const float4* xc = (const float4*)(xb + c * 16);
        float dot = 0.f;
#pragma unroll
        for (int i = 0; i < 4; i++) {
            float4 xv = xc[i];
            dot += q[4 * i + 0] * xv.x + q[4 * i + 1] * xv.y
                 + q[4 * i + 2] * xv.z + q[4 * i + 3] * xv.w;
        }
        h = rn * h + dot;
    }
}

// ---------------------------------------------------------------------------
// Kernel D: one wave per (d, group of FOUR 16-chunk column tiles).
//   Y(16x16) = P_d(16x16) * H_tile(16x16) + M_d(16x16) * X_tile(16x16)
// 8x V_WMMA_F32_16X16X4_F32 per tile; M_d/P_d A-slices are loaded ONCE per wave
// and reused across the 4 tiles (4x less A-table L2 traffic, amortized launch).
// B-matrix (4x16 f32) lane layout mirrors the documented A layout:
//   lane l: col = l&15, K-slice element = 2*(l>>4)+v  -> one b64 load per slice.
// Y is written IN PLACE over the X tile (each 256-float region owned by 1 wave).
__global__ void __launch_bounds__(128)
chunk_gemm_kernel(const float* xin,                // == yout (aliased, ws)
                  const float* __restrict__ H,
                  const float* __restrict__ MA,
                  const float* __restrict__ PA,
                  float*       yout)
{
    int lane = threadIdx.x & 31;
    int wave = threadIdx.x >> 5;
    int d  = blockIdx.y;
    int g  = blockIdx.x * 4 + wave;                // 0..15: group of 4 tiles
    int cb0 = g * 4;                               // first tile; groups never
    int b   = cb0 >> 4;                            //   straddle a batch boundary
    int cc0 = cb0 & 15;
    size_t base0 = (size_t)(d * BATCH + b) * SEQL + (size_t)cc0 * 256;
    int lrow = lane & 15, lhi = lane >> 4;
    int boff = lrow * 16 + 2 * lhi;                // B-matrix per-lane base offset

    // Hoisted A-matrix K-slices (M_d and P_d), reused for all 4 tiles.
    v2f am[4], ap[4];
#pragma unroll
    for (int k = 0; k < 4; k++) {
        size_t ai = ((size_t)(d * 4 + k) * 32 + lane) * 2;
        ap[k] = *(const v2f*)(PA + ai);
        am[k] = *(const v2f*)(MA + ai);
    }

#pragma unroll
    for (int i = 0; i < 4; i++) {
        const float* xbase = xin  + base0 + (size_t)i * 256;
        const float* hbase = H    + base0 + (size_t)i * 256;
        float*       ybase = yout + base0 + (size_t)i * 256;

        v8f acc = {0.f, 0.f, 0.f, 0.f, 0.f, 0.f, 0.f, 0.f};
#pragma unroll
        for (int k = 0; k < 4; k++) {              // carry-in term: P_d * H
            v2f bm = *(const v2f*)(hbase + boff + 4 * k);
            acc = __builtin_amdgcn_wmma_f32_16x16x4_f32(false, ap[k], false, bm,
                                                        (short)0, acc, false, false);
        }
#pragma unroll
        for (int k = 0; k < 4; k++) {              // intra-chunk term: M_d * X
            v2f bm = *(const v2f*)(xbase + boff + 4 * k);
            acc = __builtin_amdgcn_wmma_f32_16x16x4_f32(false, am[k], false, bm,
                                                        (short)0, acc, false, false);
        }
        // C/D layout: lane holds D[t = v + 8*lhi][col = lrow]; t consecutive over v
        float4 lo = make_float4(acc[0], acc[1], acc[2], acc[3]);
        float4 hi = make_float4(acc[4], acc[5], acc[6], acc[7]);
        float* yp = ybase + lrow * 16 + 8 * lhi;
        *(float4*)(yp)     = lo;
        *(float4*)(yp + 4) = hi;
    }
}

// ---------------------------------------------------------------------------
extern "C" void kernel_launch(void* const* d_in, const int* in_sizes, int n_in,
                              void* d_out, int out_size, void* d_ws, size_t ws_size,
                              hipStream_t stream)
{
    const float* x     = (const float*)d_in[0];   // (B,L,D)
    const float* A     = (const float*)d_in[1];   // (D,N)
    const float* Bm    = (const float*)d_in[2];   // (D,N)
    const float* delta = (const float*)d_in[3];   // (D,)
    float* out = (float*)d_out;                   // (B,L,D)

    float* ws = (float*)d_ws;
    float* XT = ws;                               // 8,388,608 f  (also YT, in place)
    float* H  = XT + (size_t)DCH * BT;            // 8,388,608 f
    float* MA = H  + (size_t)DCH * BT;            // 131,072 f
    float* PA = MA + DCH * 4 * 32 * 2;            // 131,072 f
    float* Q  = PA + DCH * 4 * 32 * 2;            // 131,072 f
    float* r  = Q  + DCH * 256;                   // 8,192 f      (~66 MB total)

    precompute_kernel<<<dim3(2), dim3(256), 0, stream>>>(A, Bm, delta, MA, PA, Q, r);

    // x(bt, d) -> XT(d, bt)
    transpose_kernel<<<dim3(DCH / 32, BT / 32), dim3(32, 8), 0, stream>>>(x, XT, BT, DCH);

    chunk_states_kernel<<<dim3(128), dim3(256), 0, stream>>>(XT, Q, r, H);

    chunk_gemm_kernel<<<dim3(4, DCH), dim3(128), 0, stream>>>(XT, H, MA, PA, XT);

    // YT(d, bt) -> y(bt, d)
    transpose_kernel<<<dim3(BT / 32, DCH / 32), dim3(32, 8), 0, stream>>>(XT, out, DCH, BT);
}